// ProbAttention_8933531976028
// MI455X (gfx1250) — compile-verified
//
#include <hip/hip_runtime.h>

typedef __attribute__((ext_vector_type(2))) float v2f;
typedef __attribute__((ext_vector_type(8))) float v8f;

#define B_    32
#define H_    8
#define L_    2048
#define S_    2048
#define D_    64
#define NTOP  10
#define NEGBIG 3.4e38f

// ---------------------------------------------------------------------------
// Kernel 1: M[bh][l] = max_j(Q[l]·Ksamp[j]) - sum_j(Q[l]·Ksamp[j]) / S
// ---------------------------------------------------------------------------
__global__ __launch_bounds__(128) void k_scores_M(
    const float* __restrict__ Q, const float* __restrict__ K,
    const int* __restrict__ idx_k, float* __restrict__ Mout) {
  __shared__ float ksamp[NTOP * D_];
  __shared__ float qtile[128 * (D_ + 1)];   // +1 pad: banks differ per row
  const int bh   = blockIdx.x >> 4;         // 2048/128 = 16 tiles per bh
  const int lblk = blockIdx.x & 15;
  const int tid  = threadIdx.x;

  const float* Kbh = K + (size_t)bh * L_ * D_;
  const float* Qt  = Q + (size_t)bh * L_ * D_ + (size_t)lblk * 128 * D_;

  for (int i = tid; i < NTOP * D_; i += 128) {
    int row = i >> 6, col = i & 63;
    ksamp[i] = Kbh[(size_t)idx_k[row] * D_ + col];
  }
  for (int i = tid; i < 128 * D_; i += 128) {
    int row = i >> 6, col = i & 63;
    qtile[row * (D_ + 1) + col] = Qt[i];    // coalesced global read
  }
  __syncthreads();

  const float* qr = &qtile[tid * (D_ + 1)];
  float mx = -NEGBIG, sm = 0.f;
  for (int j = 0; j < NTOP; ++j) {
    float acc = 0.f;
    #pragma unroll
    for (int d = 0; d < D_; ++d) acc += qr[d] * ksamp[j * D_ + d];
    mx = fmaxf(mx, acc);
    sm += acc;
  }
  Mout[(size_t)bh * L_ + lblk * 128 + tid] = mx - sm / (float)S_;
}

// ---------------------------------------------------------------------------
// Kernel 2: full bitonic argsort of M per (b,h): descending value, ascending
// index on ties (matches jax.lax.top_k stability).
// ---------------------------------------------------------------------------
__global__ __launch_bounds__(1024) void k_sort(
    const float* __restrict__ Mv, int* __restrict__ idx_out) {
  __shared__ float sv[L_];
  __shared__ int   si[L_];
  const int bh  = blockIdx.x;
  const int tid = threadIdx.x;
  const float* m = Mv + (size_t)bh * L_;

  sv[tid]        = m[tid];        si[tid]        = tid;
  sv[tid + 1024] = m[tid + 1024]; si[tid + 1024] = tid + 1024;
  __syncthreads();

  for (int k = 2; k <= L_; k <<= 1) {
    for (int j = k >> 1; j > 0; j >>= 1) {
      int i   = ((tid & ~(j - 1)) << 1) | (tid & (j - 1));
      int ixj = i + j;
      bool dir = ((i & k) == 0);
      float va = sv[i], vb = sv[ixj];
      int   ia = si[i], ib = si[ixj];
      // "ixj belongs before i" in (desc value, asc index) order:
      bool bef = (vb > va) || (vb == va && ib < ia);
      if (bef == dir) { sv[i] = vb; sv[ixj] = va; si[i] = ib; si[ixj] = ia; }
      __syncthreads();
    }
  }
  idx_out[(size_t)bh * L_ + tid]        = si[tid];
  idx_out[(size_t)bh * L_ + tid + 1024] = si[tid + 1024];
}

// ---------------------------------------------------------------------------
// Kernel 3: part2 — out rows 10..L-1 get cumsum-over-D of V at last_bott rows
// ---------------------------------------------------------------------------
__global__ __launch_bounds__(256) void k_part2(
    const float* __restrict__ V, const int* __restrict__ sidx,
    float* __restrict__ out) {
  long gid = (long)blockIdx.x * blockDim.x + threadIdx.x;
  const long total = (long)B_ * H_ * (L_ - NTOP);
  if (gid >= total) return;
  int bh = (int)(gid / (L_ - NTOP));
  int n  = NTOP + (int)(gid % (L_ - NTOP));
  int src = sidx[(size_t)bh * L_ + n];
  const float4* v4 = (const float4*)(V + ((size_t)bh * L_ + src) * D_);
  float4*       o4 = (float4*)(out + ((size_t)bh * L_ + n) * D_);
  float run = 0.f;
  #pragma unroll
  for (int d4 = 0; d4 < D_ / 4; ++d4) {
    float4 x = v4[d4];
    float4 o;
    o.x = run + x.x; o.y = o.x + x.y; o.z = o.y + x.z; o.w = o.z + x.w;
    run = o.w;
    o4[d4] = o;
  }
}

// ---------------------------------------------------------------------------
// Kernel 4: part1 — flash attention for the 10 selected queries, f32 WMMA.
// 4 waves per (b,h); wave w handles S-tiles w, w+4, ... (32 tiles each).
// ---------------------------------------------------------------------------
__global__ __launch_bounds__(128) void k_part1(
    const float* __restrict__ Q, const float* __restrict__ K,
    const float* __restrict__ V, const int* __restrict__ sidx,
    float* __restrict__ out) {
  __shared__ float lds_p[4 * 256];          // per-wave 16x16 prob tile
  __shared__ float lds_acc[4 * 16 * D_];    // per-wave partial part1
  __shared__ float lds_max[4 * 16];
  __shared__ float lds_sum[4 * 16];
  __shared__ int   mtop[16];

  const int bh   = blockIdx.x;
  const int tid  = threadIdx.x;
  const int wave = tid >> 5;
  const int lane = tid & 31;
  const int half = lane >> 4;               // wave32: two 16-lane halves
  const int ln   = lane & 15;

  if (tid < 16) mtop[tid] = (tid < NTOP) ? sidx[(size_t)bh * L_ + tid] : -1;
  __syncthreads();

  const float* Qbh = Q + (size_t)bh * L_ * D_;
  const float* Kbh = K + (size_t)bh * L_ * D_;
  const float* Vbh = V + (size_t)bh * L_ * D_;

  // A fragments of Q_reduce (16x64), 16 chunks of K=4.
  // A layout (16x4 f32): lane = m + 16*(k/2), reg = k%2.
  v2f aq[16];
  {
    int m    = ln;
    int qrow = (m < NTOP) ? mtop[m] : 0;    // padding rows read row 0 (masked later)
    const float* qr = Qbh + (size_t)qrow * D_;
    #pragma unroll
    for (int kc = 0; kc < 16; ++kc) {
      int dbase = kc * 4 + 2 * half;
      v2f a; a.x = qr[dbase]; a.y = qr[dbase + 1];
      aq[kc] = a;
    }
  }

  float run_max[8], run_sum[8];
  v8f acc[4];
  #pragma unroll
  for (int r = 0; r < 8; ++r) { run_max[r] = -__builtin_inff(); run_sum[r] = 0.f; }
  #pragma unroll
  for (int dt = 0; dt < 4; ++dt)
    #pragma unroll
    for (int r = 0; r < 8; ++r) acc[dt][r] = 0.f;

  const float SCALE = 0.125f;               // 1/sqrt(64)

  for (int ti = wave; ti < S_ / 16; ti += 4) {   // exactly 32 iters per wave
    const int s0 = ti * 16;

    // ---- scores tile: c = Q_reduce(16x64) @ K_tile^T(64x16) -------------
    v8f c;
    #pragma unroll
    for (int r = 0; r < 8; ++r) c[r] = 0.f;
    // B layout (4x16 f32): lane = n + 16*(k/2), reg = k%2  -> float2 load
    const float* kbase = Kbh + (size_t)(s0 + ln) * D_ + 2 * half;
    #pragma unroll
    for (int kc = 0; kc < 16; ++kc) {
      v2f b; b.x = kbase[kc * 4]; b.y = kbase[kc * 4 + 1];
      c = __builtin_amdgcn_wmma_f32_16x16x4_f32(
              false, aq[kc], false, b, (short)0, c, false, false);
    }

    // ---- scale + causal mask + online softmax update --------------------
    float p[8], alpha[8];
    #pragma unroll
    for (int r = 0; r < 8; ++r) {
      const int m = r + 8 * half;           // C/D layout row
      const int s = s0 + ln;                // C/D layout col
      float sc = c[r] * SCALE;
      if (s > mtop[m]) sc = -NEGBIG;        // mask (padding rows: always)
      float rm = sc;                        // row max across 16-lane half
      rm = fmaxf(rm, __shfl_xor(rm, 1));
      rm = fmaxf(rm, __shfl_xor(rm, 2));
      rm = fmaxf(rm, __shfl_xor(rm, 4));
      rm = fmaxf(rm, __shfl_xor(rm, 8));
      float nm = fmaxf(run_max[r], rm);
      alpha[r] = __expf(run_max[r] - nm);
      run_max[r] = nm;
      float pv = __expf(sc - nm);           // masked -> underflow to 0
      p[r] = pv;
      float ps = pv;
      ps += __shfl_xor(ps, 1);
      ps += __shfl_xor(ps, 2);
      ps += __shfl_xor(ps, 4);
      ps += __shfl_xor(ps, 8);
      run_sum[r] = run_sum[r] * alpha[r] + ps;
    }
    #pragma unroll
    for (int dt = 0; dt < 4; ++dt)
      #pragma unroll
      for (int r = 0; r < 8; ++r) acc[dt][r] *= alpha[r];

    // ---- transpose p (C-layout) -> row-major tile in LDS ----------------
    float* pt = &lds_p[wave * 256];
    #pragma unroll
    for (int r = 0; r < 8; ++r) pt[(r + 8 * half) * 16 + ln] = p[r];
    __syncthreads();

    // ---- acc += p(16x16) @ V_tile(16x64) --------------------------------
    #pragma unroll
    for (int kc = 0; kc < 4; ++kc) {
      v2f ap;                               // A layout: m=ln, k=kc*4+2*half+r
      ap.x = pt[ln * 16 + kc * 4 + 2 * half];
      ap.y = pt[ln * 16 + kc * 4 + 2 * half + 1];
      const int srow = s0 + kc * 4 + 2 * half;
      const float* v0 = Vbh + (size_t)srow * D_ + ln;
      const float* v1 = v0 + D_;
      #pragma unroll
      for (int dt = 0; dt < 4; ++dt) {
        v2f bv; bv.x = v0[dt * 16]; bv.y = v1[dt * 16];
        acc[dt] = __builtin_amdgcn_wmma_f32_16x16x4_f32(
                      false, ap, false, bv, (short)0, acc[dt], false, false);
      }
    }
    __syncthreads();
  }

  // ---- dump per-wave partial softmax state + accumulators ---------------
  #pragma unroll
  for (int r = 0; r < 8; ++r) {
    if (ln == 0) {
      lds_max[wave * 16 + r + 8 * half] = run_max[r];
      lds_sum[wave * 16 + r + 8 * half] = run_sum[r];
    }
  }
  #pragma unroll
  for (int dt = 0; dt < 4; ++dt)
    #pragma unroll
    for (int r = 0; r < 8; ++r)
      lds_acc[(wave * 16 + r + 8 * half) * D_ + dt * 16 + ln] = acc[dt][r];
  __syncthreads();

  // ---- merge the 4 waves' partial states, write rows 0..9 ---------------
  for (int e = tid; e < NTOP * D_; e += 128) {
    int m = e >> 6, d = e & 63;
    float gmax = -__builtin_inff();
    for (int w = 0; w < 4; ++w) gmax = fmaxf(gmax, lds_max[w * 16 + m]);
    float tsum = 0.f, val = 0.f;
    for (int w = 0; w < 4; ++w) {
      float f = __expf(lds_max[w * 16 + m] - gmax);
      tsum += lds_sum[w * 16 + m] * f;
      val  += lds_acc[(w * 16 + m) * D_ + d] * f;
    }
    out[((size_t)bh * L_ + m) * D_ + d] = val / tsum;
  }
}

// ---------------------------------------------------------------------------
extern "C" void kernel_launch(void* const* d_in, const int* in_sizes, int n_in,
                              void* d_out, int out_size, void* d_ws, size_t ws_size,
                              hipStream_t stream) {
  const float* Q     = (const float*)d_in[0];
  const float* K     = (const float*)d_in[1];
  const float* V     = (const float*)d_in[2];
  const int*   idx_k = (const int*)d_in[3];
  float* out = (float*)d_out;

  float* ws_M   = (float*)d_ws;                                   // B*H*L f32
  int*   ws_idx = (int*)((char*)d_ws + (size_t)B_ * H_ * L_ * sizeof(float));

  k_scores_M<<<B_ * H_ * (L_ / 128), 128, 0, stream>>>(Q, K, idx_k, ws_M);
  k_sort<<<B_ * H_, 1024, 0, stream>>>(ws_M, ws_idx);
  {
    long total  = (long)B_ * H_ * (L_ - NTOP);
    int  blocks = (int)((total + 255) / 256);
    k_part2<<<blocks, 256, 0, stream>>>(V, ws_idx, out);
  }
  k_part1<<<B_ * H_, 128, 0, stream>>>(Q, K, V, ws_idx, out);
}